// RoleAlignDecoder_49709951484646
// MI455X (gfx1250) — compile-verified
//
#include <hip/hip_runtime.h>
#include <hip/hip_bf16.h>
#include <math.h>

typedef float v2f __attribute__((ext_vector_type(2)));
typedef float v8f __attribute__((ext_vector_type(8)));

#define DDIM 128

// ---------------------------------------------------------------------------
// Phase 1: y = x_out @ W   (nNodes x 128) = (nNodes x 128) * (128 x 128)
// fp32 WMMA 16x16x4. 8 waves/block, each wave owns a 16-row M tile and the
// full N=128 (8 accumulators of 16x16). W lives in LDS (64 KB).
// ---------------------------------------------------------------------------
__global__ __launch_bounds__(256) void gemm_xw_kernel(
    const float* __restrict__ x_out, const float* __restrict__ W,
    float* __restrict__ y, int nNodes)
{
    __shared__ float sW[DDIM * DDIM];  // 64 KB, row-major [k][j]

    // Cooperative, vectorized LDS fill of W
    {
        const float4* src = (const float4*)W;
        float4* dst = (float4*)sW;
        const int nVec = DDIM * DDIM / 4;  // 4096
        for (int i = threadIdx.x; i < nVec; i += blockDim.x) dst[i] = src[i];
    }
    __syncthreads();

    const int lane = threadIdx.x & 31;
    const int wave = threadIdx.x >> 5;
    const int rowBase = blockIdx.x * 128 + wave * 16;
    if (rowBase >= nNodes) return;   // tail waves: nothing to do (no barriers after this)

    const int mRow = lane & 15;      // M (and B's N) index within tile
    const int half = lane >> 4;      // 0: lanes 0-15, 1: lanes 16-31

    int loadRow = rowBase + mRow;
    if (loadRow >= nNodes) loadRow = nNodes - 1;  // clamp loads; stores are guarded
    const float* aRow = x_out + (size_t)loadRow * DDIM;

    v8f acc[8] = {};

    for (int t = 0; t < DDIM / 4; ++t) {
        // ISA 16x4 f32 A layout: lanes 0-15 hold K=4t,4t+1; lanes 16-31 hold K=4t+2,4t+3
        const int kBase = 4 * t + 2 * half;
        v2f afrag;
        {
            const float2 av = *(const float2*)(aRow + kBase);  // 8B-aligned b64 load
            afrag.x = av.x;
            afrag.y = av.y;
        }
#pragma unroll
        for (int n = 0; n < 8; ++n) {
            const int col = n * 16 + mRow;
            v2f bfrag;  // mirrored 4x16 B layout: same K pair, N = col
            bfrag.x = sW[kBase * DDIM + col];
            bfrag.y = sW[(kBase + 1) * DDIM + col];
            // (neg_a, A, neg_b, B, c_mod, C, reuse_a, reuse_b)
            acc[n] = __builtin_amdgcn_wmma_f32_16x16x4_f32(
                false, afrag, false, bfrag, (short)0, acc[n], false, false);
        }
    }

    // 16x16 f32 C/D layout: VGPR v holds row (v + 8*half), column = lane%16
#pragma unroll
    for (int n = 0; n < 8; ++n) {
#pragma unroll
        for (int v = 0; v < 8; ++v) {
            const int r = rowBase + v + 8 * half;
            if (r < nNodes)
                y[(size_t)r * DDIM + n * 16 + mRow] = acc[n][v];
        }
    }
}

// ---------------------------------------------------------------------------
// Phase 2: out[e] = sigmoid(dot(y[src[e]], x_in[dst[e]]))
// One wave32 per edge: lane i loads float4 at offset 4*i of each 128-float row
// (32 lanes cover the full 512B row, coalesced), then butterfly-reduce.
// y + x_in together are ~102 MB -> resident in the 192 MB L2.
// ---------------------------------------------------------------------------
__global__ __launch_bounds__(256) void edge_score_kernel(
    const float* __restrict__ y, const float* __restrict__ x_in,
    const long long* __restrict__ edge_src, const long long* __restrict__ edge_dst,
    float* __restrict__ out, int nEdges)
{
    const int lane = threadIdx.x & 31;
    const int wavesPerBlock = blockDim.x >> 5;
    const int waveId = blockIdx.x * wavesPerBlock + (threadIdx.x >> 5);
    const int totalWaves = gridDim.x * wavesPerBlock;

    for (int e = waveId; e < nEdges; e += totalWaves) {
        const int s = (int)edge_src[e];
        const int d = (int)edge_dst[e];
        const float4 a = *(const float4*)(y    + (size_t)s * DDIM + lane * 4);
        const float4 b = *(const float4*)(x_in + (size_t)d * DDIM + lane * 4);
        float p = a.x * b.x + a.y * b.y + a.z * b.z + a.w * b.w;
        // wave32 butterfly reduction
#pragma unroll
        for (int m = 16; m >= 1; m >>= 1) p += __shfl_xor(p, m, 32);
        if (lane == 0) out[e] = 1.0f / (1.0f + __expf(-p));
    }
}

extern "C" void kernel_launch(void* const* d_in, const int* in_sizes, int n_in,
                              void* d_out, int out_size, void* d_ws, size_t ws_size,
                              hipStream_t stream) {
    // setup_inputs() order: x_in, x_out, edge_index (int64, shape [2,E]), W
    const float* x_in       = (const float*)d_in[0];
    const float* x_out      = (const float*)d_in[1];
    const long long* edges  = (const long long*)d_in[2];
    const float* W          = (const float*)d_in[3];

    const int nNodes = in_sizes[0] / DDIM;
    const int nEdges = in_sizes[2] / 2;

    const long long* edge_src = edges;            // row 0 -> indexes x_out
    const long long* edge_dst = edges + nEdges;   // row 1 -> indexes x_in

    float* y   = (float*)d_ws;     // nNodes * 128 * 4 bytes (~51.2 MB) scratch
    float* out = (float*)d_out;

    // Phase 1: WMMA GEMM, 128 rows per block
    {
        dim3 block(256);
        dim3 grid((nNodes + 127) / 128);
        gemm_xw_kernel<<<grid, block, 0, stream>>>(x_out, W, y, nNodes);
    }
    // Phase 2: edge gather + dot + sigmoid (memory bound, ~1.64 GB of L2-resident gathers)
    {
        dim3 block(256);
        dim3 grid(2048);
        edge_score_kernel<<<grid, block, 0, stream>>>(y, x_in, edge_src, edge_dst, out, nEdges);
    }
}